// BinaryMixedOp_63024350101904
// MI455X (gfx1250) — compile-verified
//
#include <hip/hip_runtime.h>
#include <math.h>

// ---- problem constants (from reference) ----
#define D_DIM   1024          // D
#define M_TOTAL 4096          // B*T = 8*512
#define NUM_OPS 8

// ---- tiling ----
#define BM 128
#define BN 128
#define BK 32
#define APAD 2                // LDS padding to break bank conflicts
#define BPAD 4

typedef __attribute__((ext_vector_type(2))) float v2f;
typedef __attribute__((ext_vector_type(8))) float v8f;
typedef __attribute__((ext_vector_type(4))) float f32x4;
typedef __attribute__((ext_vector_type(4))) int   i32x4;

// ---- CDNA5 async global->LDS path (guarded: probe via __has_builtin) ----
#define AS1 __attribute__((address_space(1)))
#define AS3 __attribute__((address_space(3)))

#if defined(__has_builtin)
#  if __has_builtin(__builtin_amdgcn_global_load_async_to_lds_b128)
#    define HAVE_ASYNC_LDS 1
#  endif
#endif
#ifndef HAVE_ASYNC_LDS
#  define HAVE_ASYNC_LDS 0
#endif

#if HAVE_ASYNC_LDS
__device__ __forceinline__ void async_copy16(const void* g, void* l) {
  // signature (from hipcc diagnostic): param0 is a pointer to a 128-bit
  // vector; pass typed AS-qualified pointers. (global src, lds dst, imm
  // offset, imm cpol)
  __builtin_amdgcn_global_load_async_to_lds_b128(
      (AS1 i32x4*)g, (AS3 i32x4*)l, 0, 0);
}
__device__ __forceinline__ void async_wait_all() {
#if defined(__has_builtin) && __has_builtin(__builtin_amdgcn_s_wait_asynccnt)
  __builtin_amdgcn_s_wait_asynccnt(0);
#else
  asm volatile("s_wait_asynccnt 0" ::: "memory");
#endif
}
#else
__device__ __forceinline__ void async_copy16(const void* g, void* l) {
  *(f32x4*)l = *(const f32x4*)g;      // sync fallback: global->VGPR->LDS
}
__device__ __forceinline__ void async_wait_all() {}
#endif

// --------------------------------------------------------------------------
// Kernel 1: Gumbel top-1 op selection.
// argmax(log_softmax(logits) + gumbel) == argmax(logits - log(-log(u)))
// --------------------------------------------------------------------------
__global__ void select_op_kernel(const float* __restrict__ logits,
                                 const float* __restrict__ u,
                                 int* __restrict__ idx_out) {
  if (threadIdx.x == 0 && blockIdx.x == 0) {
    float best = logits[0] - __logf(-__logf(u[0]));
    int bi = 0;
    #pragma unroll
    for (int i = 1; i < NUM_OPS; ++i) {
      float s = logits[i] - __logf(-__logf(u[i]));
      if (s > best) { best = s; bi = i; }
    }
    *idx_out = bi;
  }
}

// --------------------------------------------------------------------------
// Kernel 2: out[M,N] = x[M,K] @ W[idx][K,N] + b[idx][N]   (all f32)
// 256 threads = 8 waves; block tile 128x128; wave tile 32x64 (2x4 of 16x16).
// V_WMMA_F32_16X16X4_F32; double-buffered LDS stages filled by async
// global->LDS copies (ASYNCcnt) when available.
// --------------------------------------------------------------------------
__launch_bounds__(256, 2)
__global__ void moe_gemm_f32_wmma(const float* __restrict__ x,
                                  const float* __restrict__ W,
                                  const float* __restrict__ bias,
                                  const int* __restrict__ idx_ptr,
                                  float* __restrict__ out) {
  __shared__ float As[2][BM][BK + APAD];   // A tiles, K contiguous
  __shared__ float Bs[2][BK][BN + BPAD];   // B tiles, N contiguous

  const int idx = *idx_ptr;                                  // uniform
  const float* __restrict__ Wb = W    + (size_t)idx * D_DIM * D_DIM;
  const float* __restrict__ bb = bias + (size_t)idx * D_DIM;

  const int tid  = threadIdx.x;
  const int lane = tid & 31;
  const int wave = tid >> 5;           // 0..7

  const int m0g = blockIdx.x * BM;     // 0..4095 step 128
  const int n0g = blockIdx.y * BN;     // 0..1023 step 128

  // wave position inside block tile: 4 waves along M, 2 along N
  const int wm0 = (wave & 3) * 32;     // wave M offset (32 rows)
  const int wn0 = (wave >> 2) * 64;    // wave N offset (64 cols)

  const int lm = lane & 15;            // M (A) / N (B,C,D) within 16x16 tile
  const int lk = lane >> 4;            // selects K-pair (A/B), row-half (C/D)

  v8f acc[2][4] = {};                  // 2 (M) x 4 (N) tiles of 16x16 f32

  // stage one BMxBK A-tile + BKxBN B-tile into LDS buffer `buf`
  auto stage = [&](int buf, int k0) {
    // A: 128x32 floats = 1024 float4, 4 per thread
    #pragma unroll
    for (int p = 0; p < 4; ++p) {
      const int t  = tid + p * 256;    // float4 index 0..1023
      const int r  = t >> 3;           // 8 float4 per row
      const int c4 = t & 7;
      async_copy16(x + (size_t)(m0g + r) * D_DIM + (k0 + c4 * 4),
                   &As[buf][r][c4 * 4]);
    }
    // B: 32x128 floats = 1024 float4, 4 per thread
    #pragma unroll
    for (int p = 0; p < 4; ++p) {
      const int t  = tid + p * 256;
      const int r  = t >> 5;           // 32 float4 per row
      const int c4 = t & 31;
      async_copy16(Wb + (size_t)(k0 + r) * D_DIM + (n0g + c4 * 4),
                   &Bs[buf][r][c4 * 4]);
    }
  };

  stage(0, 0);                         // prologue fill of buffer 0

  int buf = 0;
  for (int k0 = 0; k0 < D_DIM; k0 += BK, buf ^= 1) {
    async_wait_all();                  // this wave's loads into `buf` done
    __syncthreads();                   // all waves' loads done; buf^1 free

    if (k0 + BK < D_DIM)
      stage(buf ^ 1, k0 + BK);         // prefetch next stage (async)

    // ---- compute on `buf`: 8 K-steps of 4, 8 WMMAs each ----
    #pragma unroll
    for (int kk = 0; kk < BK; kk += 4) {
      // A fragments (16x4 f32): lane%16 = M row, lane/16 picks K pair
      v2f afrag[2];
      #pragma unroll
      for (int tm = 0; tm < 2; ++tm) {
        const float* ap = &As[buf][wm0 + tm * 16 + lm][kk + lk * 2];
        afrag[tm].x = ap[0];
        afrag[tm].y = ap[1];
      }
      // B fragments (4x16 f32): lane%16 = N col, lane/16 picks K pair
      v2f bfrag[4];
      #pragma unroll
      for (int tn = 0; tn < 4; ++tn) {
        const int col = wn0 + tn * 16 + lm;
        bfrag[tn].x = Bs[buf][kk + lk * 2 + 0][col];
        bfrag[tn].y = Bs[buf][kk + lk * 2 + 1][col];
      }
      #pragma unroll
      for (int tm = 0; tm < 2; ++tm)
        #pragma unroll
        for (int tn = 0; tn < 4; ++tn)
          acc[tm][tn] = __builtin_amdgcn_wmma_f32_16x16x4_f32(
              false, afrag[tm], false, bfrag[tn],
              (short)0, acc[tm][tn], false, false);
    }
  }

  // ---- epilogue: bias add + store ----
  // C/D layout: lane%16 = N col; VGPR r holds row M = 8*(lane/16) + r
  #pragma unroll
  for (int tn = 0; tn < 4; ++tn) {
    const int col = n0g + wn0 + tn * 16 + lm;
    const float bv = bb[col];
    #pragma unroll
    for (int tm = 0; tm < 2; ++tm) {
      const int rbase = m0g + wm0 + tm * 16 + lk * 8;
      #pragma unroll
      for (int r = 0; r < 8; ++r) {
        out[(size_t)(rbase + r) * D_DIM + col] = acc[tm][tn][r] + bv;
      }
    }
  }
}

// --------------------------------------------------------------------------
extern "C" void kernel_launch(void* const* d_in, const int* in_sizes, int n_in,
                              void* d_out, int out_size, void* d_ws, size_t ws_size,
                              hipStream_t stream) {
  const float* x      = (const float*)d_in[0];  // [8,512,1024]
  const float* W      = (const float*)d_in[1];  // [8,1024,1024]
  const float* b      = (const float*)d_in[2];  // [8,1024]
  const float* logits = (const float*)d_in[3];  // [8]
  const float* u      = (const float*)d_in[4];  // [8]
  float* out          = (float*)d_out;          // [8,512,1024]
  int* idx            = (int*)d_ws;

  select_op_kernel<<<1, 32, 0, stream>>>(logits, u, idx);

  dim3 grid(M_TOTAL / BM, D_DIM / BN);          // (32, 8)
  moe_gemm_f32_wmma<<<grid, 256, 0, stream>>>(x, W, b, idx, out);
}